// ExpKernelFeatureMap_72335839200093
// MI455X (gfx1250) — compile-verified
//
#include <hip/hip_runtime.h>

// cos(x @ w^T + b): x[262144,64] f32, w[256,64] f32, b[256] f32 -> out[262144,256] f32
// Memory-bound (~335 MB @ 23.3 TB/s ~= 14 us). Strategy: stream x once, hold w in
// LDS (64 KB, transposed for conflict-free B-fragment reads), accumulate with exact
// fp32 WMMA (v_wmma_f32_16x16x4_f32), fused bias+cos epilogue, single output write.

#define HEAD_DIM   64
#define NUM_FEATS  256
#define ROWS_PER_BLOCK 128   // 8 waves x 16-row M-tiles
#define THREADS    256

typedef float v2f __attribute__((ext_vector_type(2)));
typedef float v8f __attribute__((ext_vector_type(8)));

__global__ __launch_bounds__(THREADS)
void cos_feature_map_wmma(const float* __restrict__ x,
                          const float* __restrict__ w,
                          const float* __restrict__ bias,
                          float* __restrict__ out)
{
    // lw[k*256 + f] = w[f][k]  (transposed) -> exactly 64 KB of LDS.
    // B-fragment reads: lanes index consecutive f => consecutive LDS banks, conflict-free.
    __shared__ float lw[HEAD_DIM * NUM_FEATS];

    const int tid  = threadIdx.x;
    const int lane = tid & 31;
    const int wave = tid >> 5;

    // ---- Stage w transposed into LDS: thread t owns feature row f = t. ----
    // Global reads are 16x b128 per thread; LDS writes hit distinct banks per lane
    // (all lanes store the same k, consecutive f).
    {
        const int f = tid;
        const float4* wr = (const float4*)(w + f * HEAD_DIM);
        #pragma unroll
        for (int i = 0; i < HEAD_DIM / 4; ++i) {
            float4 v = wr[i];
            lw[(4 * i + 0) * NUM_FEATS + f] = v.x;
            lw[(4 * i + 1) * NUM_FEATS + f] = v.y;
            lw[(4 * i + 2) * NUM_FEATS + f] = v.z;
            lw[(4 * i + 3) * NUM_FEATS + f] = v.w;
        }
    }
    __syncthreads();

    // ---- Per-wave 16-row M tile. Grid exactly tiles M => EXEC all-ones (WMMA req). ----
    const int m0   = blockIdx.x * ROWS_PER_BLOCK + wave * 16;
    const int mrow = lane & 15;        // A: M = lane&15 ; C/D col = n0 + (lane&15)
    const int khi  = (lane >> 4) * 2;  // A/B: K sub-offset (+2 for upper half-wave)
    const int hhi  = (lane >> 4) * 8;  // C/D: row offset (+8 for upper half-wave)

    // A fragments: 16 chunks of K=4, fp32 layout => per-lane b64 of x[m0+mrow, k0+khi .. +1].
    // x tile (4 KB/wave) is read once, reused across all 16 N-tiles.
    v2f a[16];
    const float* xrow = x + (m0 + mrow) * HEAD_DIM + khi;
    #pragma unroll
    for (int j = 0; j < 16; ++j)
        a[j] = *(const v2f*)(xrow + 4 * j);

    for (int nt = 0; nt < 16; ++nt) {                // sweep 256 features
        const int fcol = nt * 16 + (lane & 15);
        v8f acc = {};
        #pragma unroll
        for (int j = 0; j < 16; ++j) {               // K = 64 in chunks of 4
            const int k0 = 4 * j + khi;
            v2f bb;
            bb.x = lw[(k0 + 0) * NUM_FEATS + fcol];  // ds_load_b32, conflict-free
            bb.y = lw[(k0 + 1) * NUM_FEATS + fcol];
            // D = A(16x4) * B(4x16) + C, exact fp32
            acc = __builtin_amdgcn_wmma_f32_16x16x4_f32(
                      /*neg_a=*/false, a[j], /*neg_b=*/false, bb,
                      /*c_mod=*/(short)0, acc, /*reuse_a=*/false, /*reuse_b=*/false);
        }

        // Fused epilogue: bias + cos, single output write.
        const float bv = bias[fcol];                 // L2-resident broadcast read
        #pragma unroll
        for (int r = 0; r < 8; ++r) {
            const int row = m0 + r + hhi;            // fp32 C/D VGPR layout
            out[row * NUM_FEATS + fcol] = __cosf(acc[r] + bv);
        }
    }
}

extern "C" void kernel_launch(void* const* d_in, const int* in_sizes, int n_in,
                              void* d_out, int out_size, void* d_ws, size_t ws_size,
                              hipStream_t stream) {
    const float* x = (const float*)d_in[0];   // [4,16,4096,64]
    const float* w = (const float*)d_in[1];   // [256,64]
    const float* b = (const float*)d_in[2];   // [256]
    float* out = (float*)d_out;               // [4,16,4096,256]

    const int M = 4 * 16 * 4096;              // 262144 rows
    dim3 grid(M / ROWS_PER_BLOCK);            // 2048 blocks, exact tiling
    hipLaunchKernelGGL(cos_feature_map_wmma, grid, dim3(THREADS), 0, stream,
                       x, w, b, out);
}